// mdGAT_171798691917
// MI455X (gfx1250) — compile-verified
//
#include <hip/hip_runtime.h>
#include <math.h>

// ---------------------------------------------------------------------------
// GAT 3-layer pipeline for MI455X (gfx1250, wave32).
// GEMMs: V_WMMA_F32_16X16X4_F32 with LDS-staged, double-buffered tiles.
// Edge softmax/aggregation: L2-resident atomics (h matrices fit in 192MB L2).
// ---------------------------------------------------------------------------

typedef __attribute__((ext_vector_type(2))) float v2f;
typedef __attribute__((ext_vector_type(8))) float v8f;

#define NODES 10000
#define EDGES 160000
#define NEG_SLOPE 0.2f

// LDS tile geometry
#define KC   16          // K-chunk per stage
#define APAD 20          // row pitch (dwords) for A tile rows / B tile cols
#define MBLK 128         // macro-tile rows  (8 waves x 16)
#define NBLK 64          // macro-tile cols  (4 x 16 per wave)

__device__ __forceinline__ float elu_f(float x) {
    return x > 0.f ? x : (expf(x) - 1.f);
}
__device__ __forceinline__ float lrelu_f(float x) {
    return x >= 0.f ? x : NEG_SLOPE * x;
}
// Order-preserving float <-> uint mapping for atomicMax on floats.
__device__ __forceinline__ unsigned fenc(float f) {
    unsigned u = __float_as_uint(f);
    return (u & 0x80000000u) ? ~u : (u | 0x80000000u);
}
__device__ __forceinline__ float fdec(unsigned k) {
    unsigned u = (k & 0x80000000u) ? (k & 0x7FFFFFFFu) : ~k;
    return __uint_as_float(u);
}

// ---------------------------------------------------------------------------
// WMMA f32 GEMM: C[M,Nn] = A[M,K] @ B[K,Nn]  (+bias, +ELU optionally)
// 256 threads = 8 waves; macro tile MBLK x NBLK; wave w owns rows w*16..w*16+15.
// K%16==0, Nn%64==0. M arbitrary (row-clamped loads, guarded stores).
// A fragment (16x4 f32): lanes 0-15 hold K=k0,k0+1 for row=lane;
// lanes 16-31 hold K=k0+2,k0+3 for row=lane-16. B mirrors with N=lane&15
// (read from LDS B^T tile).  C/D: VGPR v -> M = v + 8*(lane>>4), N = lane&15.
// ---------------------------------------------------------------------------
__global__ __launch_bounds__(256)
void gemm_wmma_f32(const float* __restrict__ A, const float* __restrict__ B,
                   float* __restrict__ C, const float* __restrict__ bias,
                   int M, int K, int Nn, int nTiles, int act) {
    __shared__ float aT[2][MBLK * APAD];   // [row][k] pitch APAD
    __shared__ float bT[2][NBLK * APAD];   // [col][k] pitch APAD (transposed)

    const int bid = blockIdx.x;
    const int mB = bid / nTiles, nB = bid % nTiles;
    const int m0 = mB * MBLK, n0 = nB * NBLK;

    const int t    = threadIdx.x;           // 0..255
    const int w    = t >> 5;                // wave 0..7
    const int lane = t & 31;
    const int r    = lane & 15;             // row (A) / col (B) in 16x16 tile
    const int kh   = (lane >> 4) * 2;       // K sub-offset 0 or 2

    // Global-load coordinates for staging (per thread)
    const int arow0 = t >> 2;               // 0..63   (first A row)
    const int arow1 = arow0 + 64;           // 64..127 (second A row)
    const int aseg  = (t & 3) * 4;          // K sub-segment (float4)
    const int brow  = t >> 4;               // 0..15   (B K-row)
    const int bcs   = (t & 15) * 4;         // B col segment (float4)

    const int ga0 = min(m0 + arow0, M - 1);
    const int ga1 = min(m0 + arow1, M - 1);

    v8f acc0 = {}, acc1 = {}, acc2 = {}, acc3 = {};

    const int chunks = K / KC;

    // --- stage chunk 0 ---
    float4 va0 = *(const float4*)(A + (size_t)ga0 * K + aseg);
    float4 va1 = *(const float4*)(A + (size_t)ga1 * K + aseg);
    float4 vb  = *(const float4*)(B + (size_t)brow * Nn + n0 + bcs);
    *(float4*)&aT[0][arow0 * APAD + aseg] = va0;
    *(float4*)&aT[0][arow1 * APAD + aseg] = va1;
    bT[0][(bcs + 0) * APAD + brow] = vb.x;
    bT[0][(bcs + 1) * APAD + brow] = vb.y;
    bT[0][(bcs + 2) * APAD + brow] = vb.z;
    bT[0][(bcs + 3) * APAD + brow] = vb.w;
    __syncthreads();

    const int lr = w * 16 + r;              // this wave's A row in the macro tile

    for (int c = 0; c < chunks; ++c) {
        const int cur = c & 1, nxt = cur ^ 1;

        // issue global loads for next chunk (latency overlapped with compute)
        if (c + 1 < chunks) {
            const int k0 = (c + 1) * KC;
            va0 = *(const float4*)(A + (size_t)ga0 * K + k0 + aseg);
            va1 = *(const float4*)(A + (size_t)ga1 * K + k0 + aseg);
            vb  = *(const float4*)(B + (size_t)(k0 + brow) * Nn + n0 + bcs);
        }

        // compute on current buffer: 4 K-steps x 4 N-tiles
        const float* __restrict__ ap = &aT[cur][lr * APAD + kh];
        const float* __restrict__ bp = &bT[cur][r  * APAD + kh];
        #pragma unroll
        for (int kk = 0; kk < KC; kk += 4) {
            v2f a = *(const v2f*)(ap + kk);
            v2f b0 = *(const v2f*)(bp + kk);
            acc0 = __builtin_amdgcn_wmma_f32_16x16x4_f32(false, a, false, b0,
                                                         (short)0, acc0, false, false);
            v2f b1 = *(const v2f*)(bp + 16 * APAD + kk);
            acc1 = __builtin_amdgcn_wmma_f32_16x16x4_f32(false, a, false, b1,
                                                         (short)0, acc1, false, false);
            v2f b2 = *(const v2f*)(bp + 32 * APAD + kk);
            acc2 = __builtin_amdgcn_wmma_f32_16x16x4_f32(false, a, false, b2,
                                                         (short)0, acc2, false, false);
            v2f b3 = *(const v2f*)(bp + 48 * APAD + kk);
            acc3 = __builtin_amdgcn_wmma_f32_16x16x4_f32(false, a, false, b3,
                                                         (short)0, acc3, false, false);
        }

        // stage next chunk into the other buffer
        if (c + 1 < chunks) {
            *(float4*)&aT[nxt][arow0 * APAD + aseg] = va0;
            *(float4*)&aT[nxt][arow1 * APAD + aseg] = va1;
            bT[nxt][(bcs + 0) * APAD + brow] = vb.x;
            bT[nxt][(bcs + 1) * APAD + brow] = vb.y;
            bT[nxt][(bcs + 2) * APAD + brow] = vb.z;
            bT[nxt][(bcs + 3) * APAD + brow] = vb.w;
        }
        __syncthreads();
    }

    // store C (guarded rows; bias/ELU epilogue)
    const int ro = (lane >> 4) * 8;
    #pragma unroll
    for (int v = 0; v < 8; ++v) {
        const int row = m0 + w * 16 + ro + v;
        if (row >= M) continue;
        float* __restrict__ Crow = C + (size_t)row * Nn;
        float vals[4] = {acc0[v], acc1[v], acc2[v], acc3[v]};
        #pragma unroll
        for (int j = 0; j < 4; ++j) {
            int cidx = n0 + j * 16 + r;
            float x = vals[j];
            if (bias) x += bias[cidx];
            if (act)  x = elu_f(x);
            Crow[cidx] = x;
        }
    }
}

// el[n,h] = sum_d h[n,h,d]*al[h,d] ; er likewise. One wave per (n,h).
__global__ __launch_bounds__(256)
void attn_logits(const float* __restrict__ h, const float* __restrict__ al,
                 const float* __restrict__ ar, float* __restrict__ el,
                 float* __restrict__ er, int H, int D) {
    int w = blockIdx.x * (blockDim.x >> 5) + (threadIdx.x >> 5);
    int lane = threadIdx.x & 31;
    if (w >= NODES * H) return;
    int n = w / H, hh = w % H;
    const float* hp  = h  + (size_t)n * H * D + (size_t)hh * D;
    const float* alp = al + (size_t)hh * D;
    const float* arp = ar + (size_t)hh * D;
    float sl = 0.f, sr = 0.f;
    for (int d = lane; d < D; d += 32) {
        float hv = hp[d];
        sl += hv * alp[d];
        sr += hv * arp[d];
    }
    for (int off = 16; off; off >>= 1) {
        sl += __shfl_xor(sl, off, 32);
        sr += __shfl_xor(sr, off, 32);
    }
    if (lane == 0) { el[w] = sl; er[w] = sr; }
}

__global__ void fill_f32(float* __restrict__ p, float v, int n) {
    int i = blockIdx.x * blockDim.x + threadIdx.x;
    if (i < n) p[i] = v;
}
__global__ void fill_u32(unsigned* __restrict__ p, unsigned v, int n) {
    int i = blockIdx.x * blockDim.x + threadIdx.x;
    if (i < n) p[i] = v;
}

// Pass 1: segment max of leaky_relu(el[src]+er[dst]) keyed by dst.
__global__ void edge_max(const int* __restrict__ src, const int* __restrict__ dst,
                         const float* __restrict__ el, const float* __restrict__ er,
                         unsigned* __restrict__ mkey, int H) {
    int i = blockIdx.x * blockDim.x + threadIdx.x;
    if (i >= EDGES * H) return;
    int e = i / H, hh = i % H;
    int s = src[e], d = dst[e];
    float z = lrelu_f(el[s * H + hh] + er[d * H + hh]);
    atomicMax(&mkey[d * H + hh], fenc(z));
}

__global__ void decode_max(const unsigned* __restrict__ mkey,
                           float* __restrict__ m, int n) {
    int i = blockIdx.x * blockDim.x + threadIdx.x;
    if (i >= n) return;
    unsigned k = mkey[i];
    float f = (k == 0u) ? 0.f : fdec(k);
    if (!isfinite(f)) f = 0.f;   // empty segments -> 0, as in reference
    m[i] = f;
}

// Pass 2: ee = exp(z - m[dst]); store per-edge ee and segment-sum into ssum.
__global__ void edge_expsum(const int* __restrict__ src, const int* __restrict__ dst,
                            const float* __restrict__ el, const float* __restrict__ er,
                            const float* __restrict__ m, float* __restrict__ aexp,
                            float* __restrict__ ssum, int H) {
    int i = blockIdx.x * blockDim.x + threadIdx.x;
    if (i >= EDGES * H) return;
    int e = i / H, hh = i % H;
    int s = src[e], d = dst[e];
    float z = lrelu_f(el[s * H + hh] + er[d * H + hh]);
    float ee = expf(z - m[d * H + hh]);
    aexp[i] = ee;
    atomicAdd(&ssum[d * H + hh], ee);
}

// Pass 3: out[dst] += (ee/sum) * h[src]. One block per edge; coeff per head in LDS.
__global__ __launch_bounds__(256)
void edge_aggregate(const int* __restrict__ src, const int* __restrict__ dst,
                    const float* __restrict__ aexp, const float* __restrict__ ssum,
                    const float* __restrict__ h, float* __restrict__ out,
                    int H, int D) {
    __shared__ float coeff[16];
    int e = blockIdx.x;
    int s = src[e], d = dst[e];
    int t = threadIdx.x;
    if (t < H) {
        float sv = ssum[d * H + t];
        coeff[t] = aexp[e * H + t] / (sv > 0.f ? sv : 1.f);
    }
    __syncthreads();
    int F = H * D;
    const float* __restrict__ hs = h + (size_t)s * F;
    float* __restrict__ od = out + (size_t)d * F;
    for (int f = t; f < F; f += blockDim.x) {
        atomicAdd(&od[f], coeff[f / D] * hs[f]);
    }
}

// s[n,f] = elu(out[n,f] + b[f])  (in place)
__global__ void bias_elu(float* __restrict__ x, const float* __restrict__ b,
                         int F, int n) {
    int i = blockIdx.x * blockDim.x + threadIdx.x;
    if (i >= n) return;
    x[i] = elu_f(x[i] + b[i % F]);
}

// final = fw*s3 + (1-fw)*res
__global__ void combine(const float* __restrict__ s3, const float* __restrict__ res,
                        const float* __restrict__ fw, float* __restrict__ out, int n) {
    int i = blockIdx.x * blockDim.x + threadIdx.x;
    if (i >= n) return;
    float f = fw[0];
    out[i] = f * s3[i] + (1.f - f) * res[i];
}

// ---------------------------------------------------------------------------
extern "C" void kernel_launch(void* const* d_in, const int* in_sizes, int n_in,
                              void* d_out, int out_size, void* d_ws, size_t ws_size,
                              hipStream_t stream) {
    const float* x    = (const float*)d_in[0];
    const int*   src  = (const int*)d_in[1];
    const int*   dst  = (const int*)d_in[2];
    const float* W1   = (const float*)d_in[3];
    const float* b1   = (const float*)d_in[4];
    const float* al1  = (const float*)d_in[5];
    const float* ar1  = (const float*)d_in[6];
    const float* W2   = (const float*)d_in[7];
    const float* b2   = (const float*)d_in[8];
    const float* al2  = (const float*)d_in[9];
    const float* ar2  = (const float*)d_in[10];
    const float* W3   = (const float*)d_in[11];
    const float* b3   = (const float*)d_in[12];
    const float* al3  = (const float*)d_in[13];
    const float* ar3  = (const float*)d_in[14];
    const float* Wres = (const float*)d_in[15];
    const float* bres = (const float*)d_in[16];
    const float* fw   = (const float*)d_in[17];
    float* out = (float*)d_out;

    // Workspace layout (floats)
    float* bufH = (float*)d_ws;                    // [N,1280] h per layer
    float* bufS = bufH + (size_t)NODES * 1280;     // [N,1280] aggregated / s per layer
    float* bufR = bufS + (size_t)NODES * 1280;     // [N,256]  residual
    float* el   = bufR + (size_t)NODES * 256;      // [N,10]
    float* er   = el   + (size_t)NODES * 10;       // [N,10]
    float* m    = er   + (size_t)NODES * 10;       // [N,10]
    float* ssum = m    + (size_t)NODES * 10;       // [N,10]
    unsigned* mkey = (unsigned*)(ssum + (size_t)NODES * 10); // [N,10]
    float* aexp = (float*)(mkey + (size_t)NODES * 10);       // [E,10]
    (void)ws_size; (void)n_in; (void)in_sizes; (void)out_size;

    const int MB = (NODES + MBLK - 1) / MBLK;   // 79 macro M-blocks
    const int TPB = 256;
    auto grid1 = [](int n, int b) { return (n + b - 1) / b; };

    // Residual: res = elu(x @ Wres + bres)        K=512, Nn=256
    gemm_wmma_f32<<<MB * (256/64), 256, 0, stream>>>(x, Wres, bufR, bres,
                                                     NODES, 512, 256, 256/64, 1);

    // ---------------- Layer 1: H=10, D=128, F=1280 ----------------
    gemm_wmma_f32<<<MB * (1280/64), 256, 0, stream>>>(x, W1, bufH, nullptr,
                                                      NODES, 512, 1280, 1280/64, 0);
    attn_logits<<<grid1(NODES*10, 8), 256, 0, stream>>>(bufH, al1, ar1, el, er, 10, 128);
    fill_u32<<<grid1(NODES*10, TPB), TPB, 0, stream>>>(mkey, 0u, NODES*10);
    fill_f32<<<grid1(NODES*10, TPB), TPB, 0, stream>>>(ssum, 0.f, NODES*10);
    fill_f32<<<grid1(NODES*1280, TPB), TPB, 0, stream>>>(bufS, 0.f, NODES*1280);
    edge_max<<<grid1(EDGES*10, TPB), TPB, 0, stream>>>(src, dst, el, er, mkey, 10);
    decode_max<<<grid1(NODES*10, TPB), TPB, 0, stream>>>(mkey, m, NODES*10);
    edge_expsum<<<grid1(EDGES*10, TPB), TPB, 0, stream>>>(src, dst, el, er, m, aexp, ssum, 10);
    edge_aggregate<<<EDGES, 256, 0, stream>>>(src, dst, aexp, ssum, bufH, bufS, 10, 128);
    bias_elu<<<grid1(NODES*1280, TPB), TPB, 0, stream>>>(bufS, b1, 1280, NODES*1280);

    // ---------------- Layer 2: H=10, D=64, F=640  (input s1 = bufS[N,1280]) ----
    gemm_wmma_f32<<<MB * (640/64), 256, 0, stream>>>(bufS, W2, bufH, nullptr,
                                                     NODES, 1280, 640, 640/64, 0);
    attn_logits<<<grid1(NODES*10, 8), 256, 0, stream>>>(bufH, al2, ar2, el, er, 10, 64);
    fill_u32<<<grid1(NODES*10, TPB), TPB, 0, stream>>>(mkey, 0u, NODES*10);
    fill_f32<<<grid1(NODES*10, TPB), TPB, 0, stream>>>(ssum, 0.f, NODES*10);
    fill_f32<<<grid1(NODES*640, TPB), TPB, 0, stream>>>(bufS, 0.f, NODES*640);
    edge_max<<<grid1(EDGES*10, TPB), TPB, 0, stream>>>(src, dst, el, er, mkey, 10);
    decode_max<<<grid1(NODES*10, TPB), TPB, 0, stream>>>(mkey, m, NODES*10);
    edge_expsum<<<grid1(EDGES*10, TPB), TPB, 0, stream>>>(src, dst, el, er, m, aexp, ssum, 10);
    edge_aggregate<<<EDGES, 256, 0, stream>>>(src, dst, aexp, ssum, bufH, bufS, 10, 64);
    bias_elu<<<grid1(NODES*640, TPB), TPB, 0, stream>>>(bufS, b2, 640, NODES*640);

    // ---------------- Layer 3: H=1, D=256, F=256 (input s2 = bufS[N,640]) ------
    gemm_wmma_f32<<<MB * (256/64), 256, 0, stream>>>(bufS, W3, bufH, nullptr,
                                                     NODES, 640, 256, 256/64, 0);
    attn_logits<<<grid1(NODES*1, 8), 256, 0, stream>>>(bufH, al3, ar3, el, er, 1, 256);
    fill_u32<<<grid1(NODES, TPB), TPB, 0, stream>>>(mkey, 0u, NODES);
    fill_f32<<<grid1(NODES, TPB), TPB, 0, stream>>>(ssum, 0.f, NODES);
    fill_f32<<<grid1(NODES*256, TPB), TPB, 0, stream>>>(bufS, 0.f, NODES*256);
    edge_max<<<grid1(EDGES, TPB), TPB, 0, stream>>>(src, dst, el, er, mkey, 1);
    decode_max<<<grid1(NODES, TPB), TPB, 0, stream>>>(mkey, m, NODES);
    edge_expsum<<<grid1(EDGES, TPB), TPB, 0, stream>>>(src, dst, el, er, m, aexp, ssum, 1);
    edge_aggregate<<<EDGES, 256, 0, stream>>>(src, dst, aexp, ssum, bufH, bufS, 1, 256);
    bias_elu<<<grid1(NODES*256, TPB), TPB, 0, stream>>>(bufS, b3, 256, NODES*256);

    // Final fuse: out = fw*s3 + (1-fw)*res
    combine<<<grid1(NODES*256, TPB), TPB, 0, stream>>>(bufS, bufR, fw, out, NODES*256);
}